// DigitCapsuleLayer_11879879541178
// MI455X (gfx1250) — compile-verified
//
#include <hip/hip_runtime.h>
#include <hip/hip_bf16.h>
#include <math.h>

// Problem constants (from reference)
#define NB   256            // batch
#define NN   6912           // input capsules
#define DOUT 16
#define DIN  8

// Tiling: 16 batches per wave (WMMA M), NCHUNK n's per wave.
// NCHUNK=64 -> 108 chunks * 16 batch tiles = 1728 waves = 216 blocks of 8 waves:
// enough waves to cover every SIMD32 on an MI455X-class part several times over.
#define NCHUNK 64
#define CHUNKS (NN / NCHUNK)            // 108
#define BTILES (NB / 16)                // 16
#define WAVES_TOTAL (CHUNKS * BTILES)   // 1728
#define WPB 8                           // waves per block (256 threads)
#define BLOCKS (WAVES_TOTAL / WPB)      // 216

typedef __attribute__((ext_vector_type(2))) float v2f;
typedef __attribute__((ext_vector_type(8))) float v8f;

__device__ __forceinline__ v8f wmma_f32(v2f a, v2f b, v8f c) {
    // V_WMMA_F32_16X16X4_F32 : D(16x16,f32) = A(16x4,f32) * B(4x16,f32) + C
    return __builtin_amdgcn_wmma_f32_16x16x4_f32(
        /*neg_a=*/false, a, /*neg_b=*/false, b,
        /*c_mod=*/(short)0, c, /*reuse_a=*/false, /*reuse_b=*/false);
}

__global__ void caps_zero(float* __restrict__ p, int n) {
    int i = blockIdx.x * blockDim.x + threadIdx.x;
    if (i < n) p[i] = 0.0f;
}

// Pass 1: S1[b,j,e] = sum_n u_hat[b,j,n,e] as a WMMA GEMM, accumulator carried
// across the n-chunk, atomic partial reduction at the end.
__global__ __launch_bounds__(256) void caps_pass1(const float* __restrict__ X,
                                                  const float* __restrict__ Wt,
                                                  float* __restrict__ S1) {
    int wave  = blockIdx.x * WPB + (threadIdx.x >> 5);
    int lane  = threadIdx.x & 31;
    int btile = wave % BTILES;
    int chunk = wave / BTILES;
    int b0 = btile * 16;
    int n0 = chunk * NCHUNK;
    int half = lane >> 4;       // K half (A/B), M+8 half (C/D)
    int col  = lane & 15;       // A: M (batch), B/D: N (= e)

    // A lane base: x[b0+col, n, 2*half + {0,1}]  (K = d)
    const float* xp  = X  + ((long)(b0 + col) * NN + n0) * DIN + 2 * half;
    // B lane base: W[j, n, e=col, 2*half + {0,1}]
    const float* wp0 = Wt + (((long)0 * NN + n0) * DOUT + col) * DIN + 2 * half;
    const float* wp1 = Wt + (((long)1 * NN + n0) * DOUT + col) * DIN + 2 * half;

    v8f acc0 = {}; v8f acc1 = {};
    for (int i = 0; i < NCHUNK; ++i) {
        v2f a0  = *(const v2f*)(xp);        // K = 0..3
        v2f a1  = *(const v2f*)(xp + 4);    // K = 4..7
        v2f b00 = *(const v2f*)(wp0);
        v2f b01 = *(const v2f*)(wp0 + 4);
        v2f b10 = *(const v2f*)(wp1);
        v2f b11 = *(const v2f*)(wp1 + 4);
        __builtin_prefetch(xp + 32 * DIN, 0, 0);  // speculative global_prefetch_b8
        acc0 = wmma_f32(a0, b00, acc0);
        acc0 = wmma_f32(a1, b01, acc0);
        acc1 = wmma_f32(a0, b10, acc1);
        acc1 = wmma_f32(a1, b11, acc1);
        xp  += DIN;
        wp0 += DOUT * DIN;
        wp1 += DOUT * DIN;
    }
#pragma unroll
    for (int r = 0; r < 8; ++r) {
        int b = b0 + r + 8 * half;          // D layout: VGPR r = row r / r+8
        atomicAdd(&S1[(b * 2 + 0) * DOUT + col], acc0[r]);
        atomicAdd(&S1[(b * 2 + 1) * DOUT + col], acc1[r]);
    }
}

// squash(scale * S) with the 16-wide norm done via half-wave shfl reduction.
// tid -> b = tid/32, j = (tid/16)&1, e = tid&15 (matches [B][2][16] layout).
__global__ __launch_bounds__(256) void caps_squash(const float* __restrict__ S,
                                                   float* __restrict__ V,
                                                   float scale) {
    int tid = blockIdx.x * blockDim.x + threadIdx.x;
    float s = S[tid] * scale;
    float p = s * s;
    p += __shfl_xor(p, 1, 32);
    p += __shfl_xor(p, 2, 32);
    p += __shfl_xor(p, 4, 32);
    p += __shfl_xor(p, 8, 32);
    V[tid] = (p / (1.0f + p)) * s * rsqrtf(p + 1e-9f);
}

// Pass 2: recompute u per n (WMMA, C=0), t_j = v1.u (shfl reduce over e),
// c = softmax_j(t), accumulate c*u into S2.
__global__ __launch_bounds__(256) void caps_pass2(const float* __restrict__ X,
                                                  const float* __restrict__ Wt,
                                                  const float* __restrict__ V1,
                                                  float* __restrict__ S2) {
    int wave  = blockIdx.x * WPB + (threadIdx.x >> 5);
    int lane  = threadIdx.x & 31;
    int btile = wave % BTILES;
    int chunk = wave / BTILES;
    int b0 = btile * 16;
    int n0 = chunk * NCHUNK;
    int half = lane >> 4;
    int col  = lane & 15;

    const float* xp  = X  + ((long)(b0 + col) * NN + n0) * DIN + 2 * half;
    const float* wp0 = Wt + (((long)0 * NN + n0) * DOUT + col) * DIN + 2 * half;
    const float* wp1 = Wt + (((long)1 * NN + n0) * DOUT + col) * DIN + 2 * half;

    // v1 for the 8 batches this lane's accumulator rows cover
    float va[8], vb[8];
#pragma unroll
    for (int r = 0; r < 8; ++r) {
        int b = b0 + r + 8 * half;
        va[r] = V1[(b * 2 + 0) * DOUT + col];
        vb[r] = V1[(b * 2 + 1) * DOUT + col];
    }

    v8f acc0 = {}; v8f acc1 = {};
    for (int i = 0; i < NCHUNK; ++i) {
        v2f a0  = *(const v2f*)(xp);
        v2f a1  = *(const v2f*)(xp + 4);
        v2f b00 = *(const v2f*)(wp0);
        v2f b01 = *(const v2f*)(wp0 + 4);
        v2f b10 = *(const v2f*)(wp1);
        v2f b11 = *(const v2f*)(wp1 + 4);
        __builtin_prefetch(xp + 32 * DIN, 0, 0);
        v8f z = {};
        v8f d0 = wmma_f32(a0, b00, z);
        d0     = wmma_f32(a1, b01, d0);      // u[b, j=0, e]
        v8f d1 = wmma_f32(a0, b10, z);
        d1     = wmma_f32(a1, b11, d1);      // u[b, j=1, e]
#pragma unroll
        for (int r = 0; r < 8; ++r) {
            float p0 = va[r] * d0[r];        // partial of t0[batch(r,half)]
            float p1 = vb[r] * d1[r];
            p0 += __shfl_xor(p0, 1, 32);  p1 += __shfl_xor(p1, 1, 32);
            p0 += __shfl_xor(p0, 2, 32);  p1 += __shfl_xor(p1, 2, 32);
            p0 += __shfl_xor(p0, 4, 32);  p1 += __shfl_xor(p1, 4, 32);
            p0 += __shfl_xor(p0, 8, 32);  p1 += __shfl_xor(p1, 8, 32);
            float c0 = 1.0f / (1.0f + __expf(p1 - p0));   // softmax over j
            float c1 = 1.0f - c0;
            acc0[r] += c0 * d0[r];
            acc1[r] += c1 * d1[r];
        }
        xp  += DIN;
        wp0 += DOUT * DIN;
        wp1 += DOUT * DIN;
    }
#pragma unroll
    for (int r = 0; r < 8; ++r) {
        int b = b0 + r + 8 * half;
        atomicAdd(&S2[(b * 2 + 0) * DOUT + col], acc0[r]);
        atomicAdd(&S2[(b * 2 + 1) * DOUT + col], acc1[r]);
    }
}

extern "C" void kernel_launch(void* const* d_in, const int* in_sizes, int n_in,
                              void* d_out, int out_size, void* d_ws, size_t ws_size,
                              hipStream_t stream) {
    const float* X = (const float*)d_in[0];   // [256, 6912, 8]
    const float* W = (const float*)d_in[1];   // [2, 6912, 16, 8]
    float* out = (float*)d_out;               // [256, 2, 16]

    float* S1 = (float*)d_ws;                 // 8192 f32
    float* S2 = S1 + NB * 2 * DOUT;           // 8192 f32
    float* V1 = S2 + NB * 2 * DOUT;           // 8192 f32

    caps_zero<<<(2 * NB * 2 * DOUT) / 256, 256, 0, stream>>>(S1, 2 * NB * 2 * DOUT);
    caps_pass1<<<BLOCKS, 256, 0, stream>>>(X, W, S1);
    caps_squash<<<(NB * 2 * DOUT) / 256, 256, 0, stream>>>(S1, V1, 0.5f);
    caps_pass2<<<BLOCKS, 256, 0, stream>>>(X, W, V1, S2);
    caps_squash<<<(NB * 2 * DOUT) / 256, 256, 0, stream>>>(S2, out, 1.0f);
}